// PillarQueryAndGroupV2a_17884243821424
// MI455X (gfx1250) — compile-verified
//
#include <hip/hip_runtime.h>
#include <stdint.h>

// ---------------------------------------------------------------------------
// PillarQueryAndGroupV2a for MI455X (gfx1250)
//
// Pure data-movement kernel (0 FLOPs, ~616 MB traffic -> ~26us roofline at
// 23.3 TB/s). Strategy:
//   Phase 1: async DMA (global_load_async_to_lds_b128) stages the 256x32 f32
//            feature tile into LDS with an XOR chunk swizzle (bank-conflict
//            free readback under wave32), overlapped with the irregular
//            xyz / pillar-center gathers.
//   Phase 2: each lane composes its 38-float output row in an LDS staging
//            buffer (conflict-free: gcd(38,64) lane-stride analysis).
//   Phase 3: linear, fully-coalesced b128 (or b32 fallback if the base is
//            misaligned) stores of the whole 38,912 B block.
// ---------------------------------------------------------------------------

#define TILE  256
#define CFEAT 32
#define COUT  38   // 32 features + 3 abs xyz + 3 rel-to-center

typedef int v4i __attribute__((ext_vector_type(4)));
typedef __attribute__((address_space(1))) v4i gv4i;   // global int4
typedef __attribute__((address_space(3))) v4i lv4i;   // LDS int4

__device__ __forceinline__ void async_copy16(const void* g, void* l) {
  // global -> LDS, 16 bytes per lane, tracked by ASYNCcnt
  __builtin_amdgcn_global_load_async_to_lds_b128((gv4i*)g, (lv4i*)l,
                                                 /*imm offset*/0, /*cpol*/0);
}

__global__ __launch_bounds__(TILE) void pillar_group_kernel(
    const float* __restrict__ xyz,        // (N,3)
    const float* __restrict__ feat,       // (N,32)
    const float* __restrict__ centers,    // (M,3)
    const float* __restrict__ pc_min,     // (3,)
    const int*   __restrict__ point_idx,  // (L,)
    const int*   __restrict__ pillar_idx, // (L,)
    float* __restrict__ gf,               // (L,38) output
    int L)
{
  __shared__ float s_feat[TILE * CFEAT];  // 32 KB feature tile (swizzled)
  __shared__ float s_out [TILE * COUT];   // 38 KB output staging

  const int t  = threadIdx.x;
  const int p0 = blockIdx.x * TILE;
  const int p  = p0 + t;
  const bool valid = (p < L);

  int src = 0;
  if (valid) src = point_idx[p];

  // ---- Phase 1: async DMA of this lane's 128B feature row into LDS --------
  if (valid) {
    const char* g  = (const char*)(feat + (size_t)src * CFEAT);
    char*       l  = (char*)s_feat + (size_t)t * (CFEAT * 4);
    const int   sw = (t >> 1) & 7;  // XOR swizzle -> conflict-free readback
#pragma unroll
    for (int k = 0; k < 8; ++k) {
      async_copy16(g + k * 16, l + ((k ^ sw) * 16));
    }
  }

  // ---- Overlap: irregular gathers while the DMA engine fills LDS ----------
  float x = 0.f, y = 0.f, z = 0.f, cx = 0.f, cy = 0.f, cz = 0.f;
  if (valid) {
    x = xyz[(size_t)src * 3 + 0];
    y = xyz[(size_t)src * 3 + 1];
    z = xyz[(size_t)src * 3 + 2];
    const int pj = pillar_idx[p];
    cx = centers[(size_t)pj * 3 + 0];
    cy = centers[(size_t)pj * 3 + 1];
    cz = centers[(size_t)pj * 3 + 2];
  }
  const float mx = pc_min[0], my = pc_min[1], mz = pc_min[2];

  __builtin_amdgcn_s_wait_asynccnt(0);   // drain ASYNCcnt: LDS tile is ready
  __syncthreads();

  // ---- Phase 2: compose the 38-float row in LDS ---------------------------
  if (valid) {
    float*        r  = &s_out[t * COUT];
    const float4* f4 = (const float4*)&s_feat[t * CFEAT];
    const int     sw = (t >> 1) & 7;
#pragma unroll
    for (int k = 0; k < 8; ++k) {
      const float4 v = f4[k ^ sw];       // undo swizzle
      r[k * 4 + 0] = v.x; r[k * 4 + 1] = v.y;
      r[k * 4 + 2] = v.z; r[k * 4 + 3] = v.w;
    }
    r[32] = x + mx;  r[33] = y + my;  r[34] = z + mz;   // absolute xyz
    r[35] = x - cx;  r[36] = y - cy;  r[37] = z - cz;   // rel to center
  }
  __syncthreads();

  // ---- Phase 3: coalesced linear writeback of the whole block -------------
  const int nvalid = (L - p0 < TILE) ? (L - p0) : TILE;
  const int nfl    = nvalid * COUT;
  float* dst = gf + (size_t)p0 * COUT;

  if (((((uintptr_t)dst) & 15u) == 0) && ((nfl & 3) == 0)) {
    float4*       d4 = (float4*)dst;
    const float4* s4 = (const float4*)s_out;
    for (int i = t, e = nfl >> 2; i < e; i += TILE) d4[i] = s4[i];  // b128
  } else {
    for (int i = t; i < nfl; i += TILE) dst[i] = s_out[i];          // b32
  }
}

// Passthrough outputs (pillars, pillar_set_indices) written as float values,
// matching numpy's int->float promotion when flat-concatenating the tuple.
__global__ void i2f_kernel(const int* __restrict__ a, float* __restrict__ o,
                           int n) {
  const int i = blockIdx.x * blockDim.x + threadIdx.x;
  if (i < n) o[i] = (float)a[i];
}

extern "C" void kernel_launch(void* const* d_in, const int* in_sizes, int n_in,
                              void* d_out, int out_size, void* d_ws, size_t ws_size,
                              hipStream_t stream) {
  const float* xyz     = (const float*)d_in[0];
  const float* feat    = (const float*)d_in[1];
  const float* centers = (const float*)d_in[2];
  const int*   pillars = (const int*)d_in[3];
  const float* pc_min  = (const float*)d_in[4];
  const int*   pset    = (const int*)d_in[5];
  const int*   psi     = (const int*)d_in[6];

  const int M3 = in_sizes[3];   // M*3 pillar ints
  const int L  = in_sizes[6];   // group rows (= N here)

  float* out = (float*)d_out;

  // Output layout (flat, tuple return order):
  //   [0, M3)            pillars (int -> float)
  //   [M3, M3+L)         pillar_set_indices (int -> float)
  //   [M3+L, M3+L+L*38)  group_features
  if (M3 > 0)
    i2f_kernel<<<(M3 + 255) / 256, 256, 0, stream>>>(pillars, out, M3);
  if (L > 0)
    i2f_kernel<<<(L + 255) / 256, 256, 0, stream>>>(psi, out + M3, L);

  float* gf = out + (size_t)M3 + (size_t)L;
  const int blocks = (L + TILE - 1) / TILE;
  if (blocks > 0)
    pillar_group_kernel<<<blocks, TILE, 0, stream>>>(
        xyz, feat, centers, pc_min, pset, psi, gf, L);
}